// BasicTJDLayer_3393024163845
// MI455X (gfx1250) — compile-verified
//
#include <hip/hip_runtime.h>
#include <hip/hip_bf16.h>

#define EPS 1e-10f
#define RANK 32
#define VOCAB 50257
#define BATCH 1024
#define NREP 8
#define NSLICE 16
#define VS_PER_SLICE 3142   // ceil(50257/16)

typedef float v2f __attribute__((ext_vector_type(2)));
typedef float v8f __attribute__((ext_vector_type(8)));

// ---------------------------------------------------------------------------
// Kernel 1: partial reduction of relu(core) over the vocab axis.
// core layout: core[r][v][c], flat = r*V*32 + v*32 + c.
// Grid: 512 blocks = (r in 0..31) x (slice in 0..15), 256 threads.
// Thread t reads float4 at column group c0=(4t)%32; deterministic output.
// ---------------------------------------------------------------------------
__global__ void reduce_core_partial(const float* __restrict__ core,
                                    float* __restrict__ partial) {
    __shared__ float4 s4[256];
    const int bid   = blockIdx.x;
    const int r     = bid >> 4;
    const int slice = bid & 15;
    const int v0 = slice * VS_PER_SLICE;
    const int v1 = min(VOCAB, v0 + VS_PER_SLICE);
    const size_t base = (size_t)r * VOCAB * RANK;
    const int t = threadIdx.x;

    float4 acc = make_float4(0.f, 0.f, 0.f, 0.f);
    const int end = v1 * RANK;
    for (int idx = v0 * RANK + t * 4; idx < end; idx += 1024) {
        const float4 x = *(const float4*)(core + base + idx);
        acc.x += fmaxf(x.x, 0.f);
        acc.y += fmaxf(x.y, 0.f);
        acc.z += fmaxf(x.z, 0.f);
        acc.w += fmaxf(x.w, 0.f);
    }
    s4[t] = acc;
    __syncthreads();

    if (t < RANK) {
        // column c = t is covered by component (t&3) of threads t' == (t>>2) mod 8
        const int g = t >> 2, comp = t & 3;
        float s = 0.f;
        for (int k = 0; k < 32; ++k) {
            const float4 q = s4[g + 8 * k];
            s += (comp == 0) ? q.x : (comp == 1) ? q.y : (comp == 2) ? q.z : q.w;
        }
        partial[bid * RANK + t] = s;  // partial[(r*16+slice)*32 + c]
    }
}

// ---------------------------------------------------------------------------
// Kernel 2: finalize M = sum_slices(partial) + V*eps, then z = a * M^8 * b.
// One block, 1024 threads, fixed summation order (deterministic).
// ---------------------------------------------------------------------------
__global__ void finalize_M_and_z(const float* __restrict__ partial,
                                 const float* __restrict__ alpha,
                                 const float* __restrict__ beta,
                                 float* __restrict__ z_out) {
    __shared__ float Ml[RANK * RANK];
    __shared__ float u[RANK];
    const int t = threadIdx.x;
    const int i = t >> 5, j = t & 31;

    float s = 0.f;
    for (int sl = 0; sl < NSLICE; ++sl)
        s += partial[(i * NSLICE + sl) * RANK + j];
    Ml[t] = s + (float)VOCAB * EPS;       // sum(relu(x)+eps) = sum(relu) + V*eps
    if (t < RANK) u[t] = fmaxf(alpha[t], 0.f) + EPS;
    __syncthreads();

    for (int it = 0; it < NREP; ++it) {
        float nu = 0.f;
        if (t < RANK) {
            for (int ii = 0; ii < RANK; ++ii)
                nu += u[ii] * Ml[ii * RANK + t];   // u = u @ M
        }
        __syncthreads();
        if (t < RANK) u[t] = nu;
        __syncthreads();
    }
    if (t == 0) {
        float z = 0.f;
        for (int jj = 0; jj < RANK; ++jj)
            z += u[jj] * (fmaxf(beta[jj], 0.f) + EPS);
        z_out[0] = z;
    }
}

// ---------------------------------------------------------------------------
// Kernel 3: per-batch-row TT chain via V_WMMA_F32_16X16X4_F32.
// One wave (32 lanes) per batch row. w[j] = sum_i v[i] * G[i, y_t, j]:
//   A tile (16x4) = G^T chunk: A[m,k] = relu(G[4kc+k, mt*16+m])+eps
//     ISA layout: lanes 0-15 -> M=lane, K=0/1; lanes 16-31 -> M=lane-16, K=2/3
//   B tile (4x16) = v[4kc+k] replicated across all 16 columns
//   Accumulate over 8 K-chunks; two M-tiles give w[0..15], w[16..31].
// D layout (C/D 16x16 f32): VGPR v -> lanes 0-15 = M=v, lanes 16-31 = M=v+8.
// ---------------------------------------------------------------------------
__global__ void tt_chain_wmma(const float* __restrict__ core,
                              const float* __restrict__ alpha,
                              const float* __restrict__ beta,
                              const int*   __restrict__ labels,
                              float* __restrict__ pt_out) {
    __shared__ float vsh[RANK];
    const int lane = threadIdx.x;          // 0..31
    const int row  = blockIdx.x;           // batch row

    vsh[lane] = fmaxf(alpha[lane], 0.f) + EPS;
    __syncthreads();

    const int    mrow   = lane & 15;
    const int    khalf  = (lane < 16) ? 0 : 2;   // K offset within 4-chunk
    const size_t rowstr = (size_t)VOCAB * RANK;  // stride between G rows (i)

    for (int t = 0; t < NREP; ++t) {
        const int label = labels[row * NREP + t];
        const float* G = core + (size_t)label * RANK;  // + i*rowstr + j
        if (t + 1 < NREP) {
            const int nl = labels[row * NREP + t + 1];
            __builtin_prefetch(core + (size_t)nl * RANK, 0, 1); // global_prefetch_b8
        }

        v8f acc0 = {};  // M-tile 0: w[0..15]
        v8f acc1 = {};  // M-tile 1: w[16..31]
#pragma unroll
        for (int kc = 0; kc < 8; ++kc) {
            const int k0 = kc * 4 + khalf;
            v2f b;
            b.x = vsh[k0];
            b.y = vsh[k0 + 1];
            const float* g0 = G + (size_t)k0 * rowstr;
            const float* g1 = G + (size_t)(k0 + 1) * rowstr;
            v2f a0, a1;
            a0.x = fmaxf(g0[mrow],      0.f) + EPS;
            a0.y = fmaxf(g1[mrow],      0.f) + EPS;
            a1.x = fmaxf(g0[16 + mrow], 0.f) + EPS;
            a1.y = fmaxf(g1[16 + mrow], 0.f) + EPS;
            acc0 = __builtin_amdgcn_wmma_f32_16x16x4_f32(
                       false, a0, false, b, (short)0, acc0, false, false);
            acc1 = __builtin_amdgcn_wmma_f32_16x16x4_f32(
                       false, a1, false, b, (short)0, acc1, false, false);
        }
        __syncthreads();                 // everyone done reading old vsh
        if ((lane & 15) == 0) {          // lane 0 -> M=v rows, lane 16 -> M=v+8
            const int mbase = (lane < 16) ? 0 : 8;
#pragma unroll
            for (int vv = 0; vv < 8; ++vv) {
                vsh[mbase + vv]      = acc0[vv];
                vsh[16 + mbase + vv] = acc1[vv];
            }
        }
        __syncthreads();
    }

    // prob_tilde = v . (relu(beta)+eps), wave32 butterfly reduce
    float pt = vsh[lane] * (fmaxf(beta[lane], 0.f) + EPS);
    for (int o = 16; o > 0; o >>= 1) pt += __shfl_xor(pt, o, 32);
    if (lane == 0) pt_out[row] = pt;
}

// ---------------------------------------------------------------------------
// Kernel 4: prob = pt / z ; loss = -mean(log(prob + eps)).
// One block, 1024 threads.
// ---------------------------------------------------------------------------
__global__ void finalize_loss(const float* __restrict__ pt,
                              const float* __restrict__ z_in,
                              float* __restrict__ out) {
    __shared__ float red[BATCH];
    const int t = threadIdx.x;
    const float z = z_in[0];
    const float p = pt[t] / z;
    out[1 + t] = p;
    red[t] = logf(p + EPS);
    __syncthreads();
    for (int s = BATCH / 2; s > 0; s >>= 1) {
        if (t < s) red[t] += red[t + s];
        __syncthreads();
    }
    if (t == 0) out[0] = -red[0] / (float)BATCH;
}

// ---------------------------------------------------------------------------
extern "C" void kernel_launch(void* const* d_in, const int* in_sizes, int n_in,
                              void* d_out, int out_size, void* d_ws, size_t ws_size,
                              hipStream_t stream) {
    const float* alpha  = (const float*)d_in[0];   // (32,)
    const float* beta   = (const float*)d_in[1];   // (32,)
    const float* core   = (const float*)d_in[2];   // (32, 50257, 32)
    const int*   labels = (const int*)d_in[3];     // (1024, 8)
    float* out = (float*)d_out;                    // [loss, prob(1024)]

    float* ws      = (float*)d_ws;
    float* partial = ws;                 // 512*32 = 16384 floats
    float* z_buf   = ws + 16384;         // 1 float (padded)
    float* pt_buf  = ws + 16416;         // 1024 floats

    reduce_core_partial<<<dim3(32 * NSLICE), dim3(256), 0, stream>>>(core, partial);
    finalize_M_and_z<<<dim3(1), dim3(1024), 0, stream>>>(partial, alpha, beta, z_buf);
    tt_chain_wmma<<<dim3(BATCH), dim3(32), 0, stream>>>(core, alpha, beta, labels, pt_buf);
    finalize_loss<<<dim3(1), dim3(1024), 0, stream>>>(pt_buf, z_buf, out);
}